// LatentSpaceDecoder_5523327942684
// MI455X (gfx1250) — compile-verified
//
#include <hip/hip_runtime.h>
#include <math.h>

// ---------------------------------------------------------------------------
// LatentSpaceDecoder for MI455X (gfx1250, wave32)
//
// Phase 1: wz = z @ W^T + b      via V_WMMA_F32_16X16X4_F32 (full fp32)
// Phase 2: out[e] = sigmoid(dot(z[src[e]], wz[dst[e]]))  -- one wave per edge
//
// z + wz = 102.4 MB -> resident in the 192 MB L2, so the 1.64 GB of edge
// gathers are served from L2, not HBM.
// ---------------------------------------------------------------------------

typedef float v2f __attribute__((ext_vector_type(2)));
typedef float v8f __attribute__((ext_vector_type(8)));

#define DFEAT 128

// ---------------------------------------------------------------------------
// Phase 1: node-level GEMM. One wave computes a 16-node x 128-feature stripe:
// 8 accumulator tiles of 16x16, K swept in steps of 4 (f32 WMMA).
//
// A (16x4 f32) per ISA 7.12.2: lane m = lane%16; VGPR0/1 hold K = 2*(lane/16),
// 2*(lane/16)+1.  B (4x16): lane n = lane%16; VGPR0/1 hold rows
// K = 2*(lane/16), +1.  B = W^T, i.e. B[k][n] = W[n*128 + k].
// C/D (16x16 f32): VGPR v holds row M = v + 8*(lane/16), col N = lane%16.
// ---------------------------------------------------------------------------
__global__ void __launch_bounds__(256)
node_gemm_wmma_f32(const float* __restrict__ z,
                   const float* __restrict__ W,
                   const float* __restrict__ bias,
                   float* __restrict__ wz,
                   int nNodes)
{
    const int lane  = threadIdx.x & 31;
    const int wave  = (int)(blockIdx.x * (blockDim.x >> 5) + (threadIdx.x >> 5));
    const int m0    = wave * 16;
    if (m0 >= nNodes) return;

    const int half = lane >> 4;   // 0 or 1
    const int l16  = lane & 15;

    // Accumulators initialized with the bias: element (M,N) of tile j has
    // N = j*16 + l16, and bias depends only on N -> splat across all 8 VGPRs.
    v8f acc[8];
#pragma unroll
    for (int j = 0; j < 8; ++j) {
        const float bv = bias[j * 16 + l16];
        v8f t = {bv, bv, bv, bv, bv, bv, bv, bv};
        acc[j] = t;
    }

    // Clamp read row for safety on a ragged tail (N here is a multiple of 16).
    int arow = m0 + l16;
    if (arow >= nNodes) arow = nNodes - 1;
    const float* __restrict__ zrow = z + (size_t)arow * DFEAT;

    for (int k = 0; k < DFEAT; k += 4) {
        // A fragment: 2 consecutive floats of this lane's z row (8B aligned).
        const v2f a = *(const v2f*)(zrow + k + 2 * half);
#pragma unroll
        for (int j = 0; j < 8; ++j) {
            // B fragment: B[k'][n] = W[n][k'], n = j*16 + l16, k' = k + 2*half.
            const v2f bb =
                *(const v2f*)(W + (size_t)(j * 16 + l16) * DFEAT + k + 2 * half);
            acc[j] = __builtin_amdgcn_wmma_f32_16x16x4_f32(
                /*neg_a=*/false, a, /*neg_b=*/false, bb,
                /*c_mod=*/(short)0, acc[j],
                /*reuse_a=*/false, /*reuse_b=*/false);
        }
    }

    // Store: element v of tile j -> row m0 + v + 8*half, col j*16 + l16.
#pragma unroll
    for (int j = 0; j < 8; ++j) {
        float* __restrict__ outb =
            wz + (size_t)(m0 + 8 * half) * DFEAT + j * 16 + l16;
#pragma unroll
        for (int v = 0; v < 8; ++v) {
            if (m0 + 8 * half + v < nNodes)
                outb[(size_t)v * DFEAT] = acc[j][v];
        }
    }
}

// ---------------------------------------------------------------------------
// Phase 2: one wave32 per edge. Each lane loads float4 (16B) of the 512B row
// of z[src] and wz[dst] (fully coalesced, 4 cachelines per row), 4 FMAs,
// then a 5-step xor-shuffle reduction; lane 0 applies sigmoid and stores.
// ---------------------------------------------------------------------------
__global__ void __launch_bounds__(256)
edge_dot_sigmoid(const float* __restrict__ z,
                 const float* __restrict__ wz,
                 const long long* __restrict__ srcIdx,
                 const long long* __restrict__ dstIdx,
                 float* __restrict__ out,
                 int nEdges)
{
    const int lane = threadIdx.x & 31;
    const int edge = (int)((blockIdx.x * blockDim.x + threadIdx.x) >> 5);
    if (edge >= nEdges) return;

    const long long s = srcIdx[edge];
    const long long d = dstIdx[edge];

    const float4* __restrict__ pa = (const float4*)(z  + (size_t)s * DFEAT);
    const float4* __restrict__ pb = (const float4*)(wz + (size_t)d * DFEAT);

    const float4 a = pa[lane];
    const float4 b = pb[lane];

    float dot = fmaf(a.x, b.x, fmaf(a.y, b.y, fmaf(a.z, b.z, a.w * b.w)));

#pragma unroll
    for (int off = 16; off > 0; off >>= 1)
        dot += __shfl_xor(dot, off, 32);

    if (lane == 0)
        out[edge] = 1.0f / (1.0f + expf(-dot));
}

// ---------------------------------------------------------------------------
// Launcher. Inputs (setup_inputs order): z [N*128] f32, edge_index [2*E] i64,
// W [128*128] f32, b [128] f32. Output: [E] f32. Workspace: wz (N*128 f32).
// ---------------------------------------------------------------------------
extern "C" void kernel_launch(void* const* d_in, const int* in_sizes, int n_in,
                              void* d_out, int out_size, void* d_ws, size_t ws_size,
                              hipStream_t stream)
{
    const float*     z          = (const float*)d_in[0];
    const long long* edge_index = (const long long*)d_in[1];
    const float*     W          = (const float*)d_in[2];
    const float*     bias       = (const float*)d_in[3];

    float* out = (float*)d_out;
    float* wz  = (float*)d_ws;                 // N*128 floats = 51.2 MB

    const int nNodes = in_sizes[0] / DFEAT;    // 100000
    const int nEdges = in_sizes[1] / 2;        // 1600000

    // Phase 1: 1 wave per 16-node stripe; 8 waves (256 threads) per block.
    const int nWaves1  = (nNodes + 15) / 16;
    const int nBlocks1 = (nWaves1 + 7) / 8;
    node_gemm_wmma_f32<<<dim3(nBlocks1), dim3(256), 0, stream>>>(
        z, W, bias, wz, nNodes);

    // Phase 2: 1 wave per edge; 8 edges (256 threads) per block.
    const int nBlocks2 = (nEdges + 7) / 8;
    edge_dot_sigmoid<<<dim3(nBlocks2), dim3(256), 0, stream>>>(
        z, wz, edge_index, edge_index + (size_t)nEdges, out, nEdges);
}